// Cate1Classifier_74947179315205
// MI455X (gfx1250) — compile-verified
//
#include <hip/hip_runtime.h>
#include <hip/hip_bf16.h>

#define EMB_DIM 512
#define SEQ_T   50
#define SEQ_D   200
#define BATCH   1024
#define KDIM    2048   // 4*EMB_DIM
#define HD      1024
#define NCLS    10
#define NEGF    (-1e30f)
#define BN_EPS  1e-5f

typedef float v2f __attribute__((ext_vector_type(2)));
typedef float v8f __attribute__((ext_vector_type(8)));

// ---------------------------------------------------------------------------
// Kernel 1: SWEM masked avg+max pooling (gather-bound, the roofline hot spot).
// One block per batch row, one thread per embedding dim -> every token row is
// a single fully-coalesced 2KB burst. Masked tokens are skipped entirely
// (uniform per-block branch), halving gather bytes vs. the naive mask.
// ---------------------------------------------------------------------------
__global__ __launch_bounds__(EMB_DIM) void swem_pool_kernel(
    const int* __restrict__ title, const int* __restrict__ desc,
    const int* __restrict__ t_len, const int* __restrict__ d_len,
    const float* __restrict__ emb, float* __restrict__ swem)
{
    const int b = blockIdx.x;
    const int d = threadIdx.x;           // 0..511
    const int tl = t_len[b];             // 1..50
    const int dl = d_len[b];             // 1..200

    float* orow = swem + (size_t)b * KDIM;

    // title
    {
        const int* trow = title + b * SEQ_T;
        float s = 0.0f, m = NEGF;
        for (int t = 0; t < tl; ++t) {
            const int tok = trow[t];                       // uniform -> scalar load
            const float e = emb[(size_t)tok * EMB_DIM + d];
            s += e;
            m = fmaxf(m, e);
        }
        orow[d]           = s / (float)(tl > 0 ? tl : 1);
        orow[EMB_DIM + d] = m;
    }
    // desc
    {
        const int* drow = desc + b * SEQ_D;
        float s = 0.0f, m = NEGF;
        for (int t = 0; t < dl; ++t) {
            const int tok = drow[t];
            const float e = emb[(size_t)tok * EMB_DIM + d];
            s += e;
            m = fmaxf(m, e);
        }
        orow[2 * EMB_DIM + d] = s / (float)(dl > 0 ? dl : 1);
        orow[3 * EMB_DIM + d] = m;
    }
}

// ---------------------------------------------------------------------------
// Kernel 2: FC GEMM  H[1024,1024] = swem[1024,2048] @ W_fc[2048,1024] + b_fc
// Full f32 precision via V_WMMA_F32_16X16X4_F32. One wave (32 threads) per
// 16x16 output tile; 512 WMMA ops along K. Lane layout per CDNA5 ISA 7.12.2:
//   A (16x4): lanes 0-15 rows M=0..15; VGPR0 = K0|K2, VGPR1 = K1|K3 by half.
//   B (4x16): mirrored (lane half selects K pair, lane%16 selects N).
//   C/D:      VGPR v -> rows v (lanes 0-15) and v+8 (lanes 16-31).
// Operands are L2-resident (8MB each on a 192MB L2), so per-lane strided A
// reads are cheap; this GEMM is <1% of total time regardless.
// ---------------------------------------------------------------------------
__global__ __launch_bounds__(32) void fc_wmma_kernel(
    const float* __restrict__ A,     // swem [BATCH, KDIM]
    const float* __restrict__ Bm,    // W_fc [KDIM, HD] row-major
    const float* __restrict__ bias,  // [HD]
    float* __restrict__ H)           // [BATCH, HD]
{
    const int tileM = blockIdx.x * 16;
    const int tileN = blockIdx.y * 16;
    const int lane  = threadIdx.x;        // 0..31, EXEC all ones
    const int r     = lane & 15;
    const int hi    = lane >> 4;          // 0: K pair {k,k+1}, 1: {k+2,k+3}

    const float* aPtr = A  + (size_t)(tileM + r) * KDIM + 2 * hi;
    const float* bPtr = Bm + (size_t)(2 * hi) * HD + tileN + r;

    v8f c = {};
    for (int k = 0; k < KDIM; k += 4) {
        v2f a;
        a.x = aPtr[k];
        a.y = aPtr[k + 1];
        v2f bb;
        bb.x = bPtr[(size_t)k * HD];
        bb.y = bPtr[(size_t)(k + 1) * HD];
        // 8 args: (neg_a, A, neg_b, B, c_mod, C, reuse_a, reuse_b)
        c = __builtin_amdgcn_wmma_f32_16x16x4_f32(
                false, a, false, bb, (short)0, c, false, false);
    }

    const float bv = bias[tileN + r];
#pragma unroll
    for (int v = 0; v < 8; ++v) {
        const int row = tileM + v + 8 * hi;
        H[(size_t)row * HD + (tileN + r)] = c[v] + bv;
    }
}

// ---------------------------------------------------------------------------
// Kernel 3: BatchNorm batch statistics (training mode, population variance).
// Each 256-thread block owns 64 columns; threads are (64 cols x 4 row-strides)
// so every row access is a coalesced 256B burst. Grid = HD/64 = 16 blocks.
// ---------------------------------------------------------------------------
__global__ __launch_bounds__(256) void bn_stats_kernel(
    const float* __restrict__ H, float* __restrict__ mean, float* __restrict__ rstd)
{
    const int c   = threadIdx.x & 63;
    const int rg  = threadIdx.x >> 6;     // 0..3
    const int col = blockIdx.x * 64 + c;

    float s = 0.0f, q = 0.0f;
    for (int r = rg; r < BATCH; r += 4) {
        const float v = H[(size_t)r * HD + col];
        s += v;
        q += v * v;
    }

    __shared__ float ls[256];
    __shared__ float lq[256];
    ls[threadIdx.x] = s;
    lq[threadIdx.x] = q;
    __syncthreads();

    if (rg == 0) {
        const float st = ls[c] + ls[64 + c] + ls[128 + c] + ls[192 + c];
        const float qt = lq[c] + lq[64 + c] + lq[128 + c] + lq[192 + c];
        const float mu  = st * (1.0f / (float)BATCH);
        const float var = qt * (1.0f / (float)BATCH) - mu * mu;
        mean[col] = mu;
        rstd[col] = rsqrtf(var + BN_EPS);
    }
}

// ---------------------------------------------------------------------------
// Kernel 4: fused BN-normalize + ReLU + classifier GEMV (N=10 is too small
// for WMMA to matter). One block per batch row; each thread accumulates all
// 10 outputs over its strided slice of the 1024-dim row, LDS tree reduction.
// W_clf (40KB) is L0/L2 resident after the first block.
// ---------------------------------------------------------------------------
__global__ __launch_bounds__(256) void clf_kernel(
    const float* __restrict__ H, const float* __restrict__ mean,
    const float* __restrict__ rstd, const float* __restrict__ gamma,
    const float* __restrict__ beta, const float* __restrict__ Wc,
    const float* __restrict__ bc, float* __restrict__ out)
{
    const int b = blockIdx.x;
    float acc[NCLS];
#pragma unroll
    for (int c = 0; c < NCLS; ++c) acc[c] = 0.0f;

    for (int j = threadIdx.x; j < HD; j += 256) {
        const float hv = H[(size_t)b * HD + j];
        float hn = (hv - mean[j]) * rstd[j] * gamma[j] + beta[j];
        hn = fmaxf(hn, 0.0f);
        const float* w = Wc + (size_t)j * NCLS;
#pragma unroll
        for (int c = 0; c < NCLS; ++c) acc[c] += hn * w[c];
    }

    __shared__ float red[NCLS][256];   // 10KB of 320KB LDS
#pragma unroll
    for (int c = 0; c < NCLS; ++c) red[c][threadIdx.x] = acc[c];
    __syncthreads();

    for (int off = 128; off > 0; off >>= 1) {
        if (threadIdx.x < off) {
#pragma unroll
            for (int c = 0; c < NCLS; ++c)
                red[c][threadIdx.x] += red[c][threadIdx.x + off];
        }
        __syncthreads();
    }
    if (threadIdx.x < NCLS)
        out[b * NCLS + threadIdx.x] = red[threadIdx.x][0] + bc[threadIdx.x];
}

// ---------------------------------------------------------------------------
extern "C" void kernel_launch(void* const* d_in, const int* in_sizes, int n_in,
                              void* d_out, int out_size, void* d_ws, size_t ws_size,
                              hipStream_t stream)
{
    (void)in_sizes; (void)n_in; (void)out_size; (void)ws_size;

    const int*   title = (const int*)  d_in[0];
    const int*   desc  = (const int*)  d_in[1];
    const int*   t_len = (const int*)  d_in[2];
    const int*   d_len = (const int*)  d_in[3];
    const float* emb   = (const float*)d_in[4];
    const float* W_fc  = (const float*)d_in[5];
    const float* b_fc  = (const float*)d_in[6];
    const float* gamma = (const float*)d_in[7];
    const float* beta  = (const float*)d_in[8];
    const float* W_clf = (const float*)d_in[9];
    const float* b_clf = (const float*)d_in[10];
    float* out = (float*)d_out;

    // Workspace layout: swem (8MB) | H (4MB) | mean (4KB) | rstd (4KB)
    float* swem = (float*)d_ws;
    float* H    = swem + (size_t)BATCH * KDIM;
    float* mean = H    + (size_t)BATCH * HD;
    float* rstd = mean + HD;

    swem_pool_kernel<<<BATCH, EMB_DIM, 0, stream>>>(title, desc, t_len, d_len, emb, swem);
    fc_wmma_kernel<<<dim3(BATCH / 16, HD / 16), 32, 0, stream>>>(swem, W_fc, b_fc, H);
    bn_stats_kernel<<<HD / 64, 256, 0, stream>>>(H, mean, rstd);
    clf_kernel<<<BATCH, 256, 0, stream>>>(H, mean, rstd, gamma, beta, W_clf, b_clf, out);
}